// HybridSurvivalQ_v4_9509057593681
// MI455X (gfx1250) — compile-verified
//
#include <hip/hip_runtime.h>

#define NQ       7
#define NLAYERS  3
#define BATCH    65536
#define FDIM     64
#define WPB      4          // waves per block in MLP kernel
#define SC       132        // comb / f1 LDS row stride (dwords), odd-bank spread, even for b64
#define SH       68         // h LDS row stride

typedef float v2f __attribute__((ext_vector_type(2)));
typedef float v8f __attribute__((ext_vector_type(8)));

// ---------------------------------------------------------------------------
// Quantum statevector kernel: one wave32 per batch element.
// Lane L owns amplitudes k = (L<<2)|j, j=0..3  (k bits[6:2]=lane, bits[1:0]=j)
// wire w <-> bit (6-w): w in 0..4 -> lane bit (4-w); w=5 -> j bit1; w=6 -> j bit0
// ---------------------------------------------------------------------------

static __device__ __forceinline__ float2 cmulr(float2 a, float cr, float ci) {
    return make_float2(a.x * cr - a.y * ci, a.x * ci + a.y * cr);
}

static __device__ __forceinline__ void apply_ry(float2 a[4], int w, float c, float s, int lane) {
    if (w <= 4) {
        const int m = 4 - w, mask = 1 << m;
        const float t = ((lane >> m) & 1) ? s : -s;   // bit0 lane: -s*partner ; bit1 lane: +s*partner
#pragma unroll
        for (int j = 0; j < 4; ++j) {
            float px = __shfl_xor(a[j].x, mask, 32);
            float py = __shfl_xor(a[j].y, mask, 32);
            a[j].x = c * a[j].x + t * px;
            a[j].y = c * a[j].y + t * py;
        }
    } else if (w == 5) {            // pairs (0,2),(1,3)
#pragma unroll
        for (int j = 0; j < 2; ++j) {
            float2 lo = a[j], hi = a[j + 2];
            a[j]     = make_float2(c * lo.x - s * hi.x, c * lo.y - s * hi.y);
            a[j + 2] = make_float2(s * lo.x + c * hi.x, s * lo.y + c * hi.y);
        }
    } else {                        // w==6: pairs (0,1),(2,3)
#pragma unroll
        for (int j = 0; j < 4; j += 2) {
            float2 lo = a[j], hi = a[j + 1];
            a[j]     = make_float2(c * lo.x - s * hi.x, c * lo.y - s * hi.y);
            a[j + 1] = make_float2(s * lo.x + c * hi.x, s * lo.y + c * hi.y);
        }
    }
}

static __device__ __forceinline__ void apply_rz(float2 a[4], int w, float c, float s, int lane) {
    // amp bit=0: *(c - i s) ; bit=1: *(c + i s)
    if (w <= 4) {
        const float si = ((lane >> (4 - w)) & 1) ? s : -s;
#pragma unroll
        for (int j = 0; j < 4; ++j) a[j] = cmulr(a[j], c, si);
    } else if (w == 5) {
#pragma unroll
        for (int j = 0; j < 4; ++j) { float si = (j & 2) ? s : -s; a[j] = cmulr(a[j], c, si); }
    } else {
#pragma unroll
        for (int j = 0; j < 4; ++j) { float si = (j & 1) ? s : -s; a[j] = cmulr(a[j], c, si); }
    }
}

static __device__ __forceinline__ void apply_cnot(float2 a[4], int c, int t, int lane) {
    if (c <= 4 && t <= 4) {                       // both in lane bits: cross-lane swap when ctrl set
        const int cm = 1 << (4 - c), tm = 1 << (4 - t);
        const bool sel = (lane & cm) != 0;
#pragma unroll
        for (int j = 0; j < 4; ++j) {
            float px = __shfl_xor(a[j].x, tm, 32);
            float py = __shfl_xor(a[j].y, tm, 32);
            if (sel) { a[j].x = px; a[j].y = py; }
        }
    } else if (c <= 4) {                          // ctrl lane bit, tgt in j bits: intra-lane swap
        const bool sel = (lane & (1 << (4 - c))) != 0;
        if (t == 5) {
            if (sel) { float2 tp = a[0]; a[0] = a[2]; a[2] = tp; tp = a[1]; a[1] = a[3]; a[3] = tp; }
        } else {
            if (sel) { float2 tp = a[0]; a[0] = a[1]; a[1] = tp; tp = a[2]; a[2] = a[3]; a[3] = tp; }
        }
    } else if (t <= 4) {                          // ctrl j bit, tgt lane bit: shuffle ctrl=1 amps
        const int tm = 1 << (4 - t);
        const int jb = (c == 5) ? 2 : 1;
#pragma unroll
        for (int j = 0; j < 4; ++j) {
            if (j & jb) {
                a[j].x = __shfl_xor(a[j].x, tm, 32);
                a[j].y = __shfl_xor(a[j].y, tm, 32);
            }
        }
    } else {                                      // both in j bits
        if (c == 5) { float2 tp = a[2]; a[2] = a[3]; a[3] = tp; }   // (5,6): j=2<->3
        else        { float2 tp = a[1]; a[1] = a[3]; a[3] = tp; }   // (6,5): j=1<->3
    }
}

__global__ __launch_bounds__(256) void quantum_kernel(
    const float* __restrict__ x_q, const float* __restrict__ scales,
    const float* __restrict__ biases, const float* __restrict__ w_ry,
    const float* __restrict__ w_rz, float* __restrict__ q_out)
{
    const int lane = threadIdx.x & 31;
    const int row  = blockIdx.x * (blockDim.x >> 5) + (threadIdx.x >> 5);
    if (row >= BATCH) return;

    float xq[NQ], sc[NQ], bi[NQ];
#pragma unroll
    for (int i = 0; i < NQ; ++i) {
        xq[i] = x_q[row * NQ + i];
        sc[i] = scales[i];
        bi[i] = biases[i];
    }

    float2 a[4];
#pragma unroll
    for (int j = 0; j < 4; ++j) a[j] = make_float2(0.f, 0.f);
    if (lane == 0) a[0].x = 1.f;                  // |0...0>

#pragma unroll
    for (int layer = 0; layer < NLAYERS; ++layer) {
#pragma unroll
        for (int i = 0; i < NQ; ++i) {
            float th = sc[i] * xq[i] + bi[i] + w_ry[layer * NQ + i];
            float hs, hc;
            __sincosf(0.5f * th, &hs, &hc);
            apply_ry(a, i, hc, hs, lane);
            float zs, zc;
            __sincosf(0.5f * w_rz[layer * NQ + i], &zs, &zc);
            apply_rz(a, i, zc, zs, lane);
        }
        const int shift = layer + 1;
#pragma unroll
        for (int i = 0; i < NQ; ++i) apply_cnot(a, i, (i + shift) % NQ, lane);
    }

    // expectation values: z[0..6], zz[0..6]
    float o[14];
#pragma unroll
    for (int i = 0; i < 14; ++i) o[i] = 0.f;
#pragma unroll
    for (int j = 0; j < 4; ++j) {
        const int k = (lane << 2) | j;
        const float p = a[j].x * a[j].x + a[j].y * a[j].y;
        int bits[NQ];
#pragma unroll
        for (int i = 0; i < NQ; ++i) bits[i] = (k >> (6 - i)) & 1;
#pragma unroll
        for (int i = 0; i < NQ; ++i) {
            o[i]     += bits[i] ? -p : p;
            o[7 + i] += (bits[i] ^ bits[(i + 1) % NQ]) ? -p : p;
        }
    }
#pragma unroll
    for (int m = 1; m < 32; m <<= 1)
#pragma unroll
        for (int i = 0; i < 14; ++i) o[i] += __shfl_xor(o[i], m, 32);

    float v = o[0];
#pragma unroll
    for (int i = 1; i < 14; ++i) v = (lane == i) ? o[i] : v;
    if (lane < 14) q_out[row * 14 + lane] = v;
}

// ---------------------------------------------------------------------------
// Fused MLP kernel: one wave per 16-row tile, fp32 WMMA 16x16x4 end-to-end.
// A frag: lane m/m+16 -> row m, K = kbase+{0,1}/{2,3}
// B frag: lane -> col lane%16, K rows kbase+{0,1} (lo half) / {2,3} (hi half)
// C frag: lane -> col lane%16, VGPR r -> row r (lo half) / r+8 (hi half)
// ---------------------------------------------------------------------------

static __device__ __forceinline__ v8f vzero8() {
    v8f z = {0.f, 0.f, 0.f, 0.f, 0.f, 0.f, 0.f, 0.f};
    return z;
}

template <int NT, int KSTEPS, int KROWS>
static __device__ __forceinline__ void gemm_tile(
    const float* __restrict__ Arow,   // per-lane pointer to row (col = lane%16) of A source
    const float* __restrict__ W,      // weights, row-major KROWS x N
    int N, v8f* acc, int col, int khalf)
{
#pragma unroll
    for (int ks = 0; ks < KSTEPS; ++ks) {
        const int kr = ks * 4 + khalf;
        v2f A;
        A.x = Arow[kr];
        A.y = Arow[kr + 1];
#pragma unroll
        for (int n = 0; n < NT; ++n) {
            const int cidx = n * 16 + col;
            v2f Bv;
            Bv.x = (kr     < KROWS) ? W[kr * N + cidx]       : 0.f;
            Bv.y = (kr + 1 < KROWS) ? W[(kr + 1) * N + cidx] : 0.f;
            acc[n] = __builtin_amdgcn_wmma_f32_16x16x4_f32(
                false, A, false, Bv, (short)0, acc[n], false, false);
        }
    }
}

template <int NT>
static __device__ __forceinline__ void bias_relu_store(
    float* __restrict__ dst, int stride, const float* __restrict__ bias,
    const v8f* acc, int col, int mrow)
{
#pragma unroll
    for (int n = 0; n < NT; ++n) {
        const float bn = bias[n * 16 + col];
#pragma unroll
        for (int r = 0; r < 8; ++r) {
            float vv = acc[n][r] + bn;
            dst[(r + mrow) * stride + n * 16 + col] = vv > 0.f ? vv : 0.f;
        }
    }
}

__global__ __launch_bounds__(WPB * 32) void mlp_kernel(
    const float* __restrict__ xc,  const float* __restrict__ q,
    const float* __restrict__ ew1, const float* __restrict__ eb1,
    const float* __restrict__ ew2, const float* __restrict__ eb2,
    const float* __restrict__ fw1, const float* __restrict__ fb1,
    const float* __restrict__ fw2, const float* __restrict__ fb2,
    const float* __restrict__ fw3, const float* __restrict__ fb3,
    const float* __restrict__ fw4, const float* __restrict__ fb4,
    const float* __restrict__ oscale, const float* __restrict__ obias,
    float* __restrict__ out)
{
    __shared__ float smem[WPB * (16 * SC + 16 * SH)];
    const int lane  = threadIdx.x & 31;
    const int wid   = threadIdx.x >> 5;
    const int r0    = (blockIdx.x * WPB + wid) * 16;
    const int col   = lane & 15;
    const int khalf = (lane >> 4) << 1;   // 0 or 2 (A/B K offset)
    const int mrow  = (lane >> 4) << 3;   // 0 or 8 (C row offset)

    float* comb = smem + wid * (16 * SC + 16 * SH);   // 16 x 112 (stride SC), also reused for f1 (128)
    float* hbuf = comb + 16 * SC;                     // 16 x 64  (stride SH)

    // stage comb: [ q(0..13) | c_enc(14..45) later | x_c(46..109) | zero pad 110..111 ]
    for (int t = lane; t < 16 * 14; t += 32) {
        int m = t / 14, c2 = t - m * 14;
        comb[m * SC + c2] = q[(r0 + m) * 14 + c2];
    }
    for (int t = lane; t < 16 * FDIM; t += 32) {
        int m = t >> 6, c2 = t & 63;
        comb[m * SC + 46 + c2] = xc[(r0 + m) * FDIM + c2];
    }
    if (lane < 16) { comb[lane * SC + 110] = 0.f; comb[lane * SC + 111] = 0.f; }

    // encoder layer 1: [16x64] @ [64x64], A straight from global x_c
    {
        v8f acc[4] = {vzero8(), vzero8(), vzero8(), vzero8()};
        gemm_tile<4, 16, 64>(xc + (r0 + col) * FDIM, ew1, 64, acc, col, khalf);
        bias_relu_store<4>(hbuf, SH, eb1, acc, col, mrow);
    }
    // encoder layer 2: [16x64] @ [64x32] -> comb cols 14..45
    {
        v8f acc[2] = {vzero8(), vzero8()};
        gemm_tile<2, 16, 64>(hbuf + col * SH, ew2, 32, acc, col, khalf);
        bias_relu_store<2>(comb + 14, SC, eb2, acc, col, mrow);
    }
    // fusion layer 1: [16x112(pad)] @ [110x128] -> overwrite comb cols 0..127
    {
        v8f acc[8] = {vzero8(), vzero8(), vzero8(), vzero8(),
                      vzero8(), vzero8(), vzero8(), vzero8()};
        gemm_tile<8, 28, 110>(comb + col * SC, fw1, 128, acc, col, khalf);
        bias_relu_store<8>(comb, SC, fb1, acc, col, mrow);
    }
    // fusion layer 2: [16x128] @ [128x64] -> hbuf
    {
        v8f acc[4] = {vzero8(), vzero8(), vzero8(), vzero8()};
        gemm_tile<4, 32, 128>(comb + col * SC, fw2, 64, acc, col, khalf);
        bias_relu_store<4>(hbuf, SH, fb2, acc, col, mrow);
    }
    // fusion layer 3 + 4: [16x64] @ [64x32], then dot with w4 via shuffle tree
    {
        v8f acc[2] = {vzero8(), vzero8()};
        gemm_tile<2, 16, 64>(hbuf + col * SH, fw3, 32, acc, col, khalf);

        const float b3a = fb3[col],      b3b = fb3[16 + col];
        const float w4a = fw4[col],      w4b = fw4[16 + col];
        const float b4  = fb4[0];
        const float os  = oscale[0],     ob  = obias[0];

        float pr[8];
#pragma unroll
        for (int r = 0; r < 8; ++r) {
            float f0 = acc[0][r] + b3a; f0 = f0 > 0.f ? f0 : 0.f;
            float f1 = acc[1][r] + b3b; f1 = f1 > 0.f ? f1 : 0.f;
            pr[r] = f0 * w4a + f1 * w4b;
        }
        // reduce over the 16 columns within each lane-half
#pragma unroll
        for (int m = 1; m < 16; m <<= 1)
#pragma unroll
            for (int r = 0; r < 8; ++r) pr[r] += __shfl_xor(pr[r], m, 32);

        float sel = pr[0];
#pragma unroll
        for (int r = 1; r < 8; ++r) sel = (col == r) ? pr[r] : sel;
        if (col < 8) out[r0 + col + mrow] = os * (sel + b4) + ob;
    }
}

// ---------------------------------------------------------------------------

extern "C" void kernel_launch(void* const* d_in, const int* in_sizes, int n_in,
                              void* d_out, int out_size, void* d_ws, size_t ws_size,
                              hipStream_t stream) {
    (void)in_sizes; (void)n_in; (void)out_size; (void)ws_size;
    const float* x_q    = (const float*)d_in[0];
    const float* x_c    = (const float*)d_in[1];
    const float* scales = (const float*)d_in[2];
    const float* biases = (const float*)d_in[3];
    const float* w_ry   = (const float*)d_in[4];
    const float* w_rz   = (const float*)d_in[5];
    const float* ew1    = (const float*)d_in[6];
    const float* eb1    = (const float*)d_in[7];
    const float* ew2    = (const float*)d_in[8];
    const float* eb2    = (const float*)d_in[9];
    const float* fw1    = (const float*)d_in[10];
    const float* fb1    = (const float*)d_in[11];
    const float* fw2    = (const float*)d_in[12];
    const float* fb2    = (const float*)d_in[13];
    const float* fw3    = (const float*)d_in[14];
    const float* fb3    = (const float*)d_in[15];
    const float* fw4    = (const float*)d_in[16];
    const float* fb4    = (const float*)d_in[17];
    const float* osc    = (const float*)d_in[18];
    const float* obi    = (const float*)d_in[19];

    float* qbuf = (float*)d_ws;                 // B x 14 f32 (3.67 MB)

    // 8 waves per block, one wave per batch element
    quantum_kernel<<<BATCH / 8, 256, 0, stream>>>(x_q, scales, biases, w_ry, w_rz, qbuf);

    // one wave per 16-row tile, WPB waves per block
    mlp_kernel<<<BATCH / (16 * WPB), 32 * WPB, 0, stream>>>(
        x_c, qbuf, ew1, eb1, ew2, eb2, fw1, fb1, fw2, fb2, fw3, fb3, fw4, fb4,
        osc, obi, (float*)d_out);
}